// GAT_33603824124479
// MI455X (gfx1250) — compile-verified
//
#include <hip/hip_runtime.h>
#include <hip/hip_bf16.h>

// ---------------------------------------------------------------------------
// GAT (2-layer) for MI455X / gfx1250.
//   Layer1: x[ N x 500 ] @ W1[500 x 128]  (HEADS=2, HID=64)  -> WMMA f32 16x16x4
//   Layer2: h1[N x 128 ] @ W2[128 x 32 ]  (1 head)           -> WMMA f32 16x16x4
// B is pre-packed into WMMA fragment order (one b64 load per K-step per lane).
// Edge softmax via ordered-uint atomicMax + float atomicAdd scatter.
// ---------------------------------------------------------------------------

typedef float v2f __attribute__((ext_vector_type(2)));
typedef float v8f __attribute__((ext_vector_type(8)));

// ---------------- order-preserving float<->uint for atomic max --------------
__device__ __forceinline__ unsigned f2ord(float f) {
  unsigned u = __float_as_uint(f);
  return (u & 0x80000000u) ? ~u : (u | 0x80000000u);
}
__device__ __forceinline__ float ord2f(unsigned o) {
  unsigned u = (o & 0x80000000u) ? (o ^ 0x80000000u) : ~o;
  return __uint_as_float(u);
}

// ---------------- zero a range of the workspace -----------------------------
__global__ void zero_kernel(unsigned* __restrict__ p, long long count) {
  long long i = (long long)blockIdx.x * blockDim.x + threadIdx.x;
  long long stride = (long long)gridDim.x * blockDim.x;
  for (; i < count; i += stride) p[i] = 0u;
}

// ---------------- pack B[K x NCOL] into WMMA fragment order -----------------
// For k-step kk and half p (lanes 0-15 -> k%4 in {0,1}; lanes 16-31 -> {2,3}):
//   Bp[(((kk*2 + p)*NCOL) + col)*2 + q] = B[(kk*4 + p*2 + q)*NCOL + col]
// so each lane fetches its (b.x, b.y) pair with one aligned 8-byte load.
template <int K, int NCOL>
__global__ void pack_b_kernel(const float* __restrict__ B,
                              float* __restrict__ Bp) {
  int tid = blockIdx.x * blockDim.x + threadIdx.x;
  if (tid >= K * NCOL) return;
  int k = tid / NCOL;
  int col = tid - k * NCOL;
  int kk = k >> 2;
  int p = (k >> 1) & 1;
  int q = k & 1;
  Bp[(((kk * 2 + p) * NCOL) + col) * 2 + q] = B[tid];
}

// ---------------- WMMA GEMM: C[M x NCOL] = A[M x K] * Bp ------------------
// 8 waves/block; wave w -> (mtile = w / (NCOL/16), ntile = w % (NCOL/16)).
// ROWS per block = 2048 / NCOL.  A tile staged in LDS via float4, Bp from L2.
template <int K, int NCOL>
__global__ __launch_bounds__(256) void gemm_wmma_kernel(
    const float* __restrict__ A, const float* __restrict__ Bp,
    float* __restrict__ C, int M) {
  constexpr int NT = NCOL / 16;  // n-tiles per block
  constexpr int MT = 8 / NT;     // m-tiles per block
  constexpr int ROWS = MT * 16;
  constexpr int KV = K / 4;      // float4s per row
  __shared__ __align__(16) float As[ROWS * K];

  const int tid = threadIdx.x;
  const int wave = tid >> 5;
  const int lane = tid & 31;
  const int row0 = blockIdx.x * ROWS;

  // cooperative A-tile load, vectorized float4 (rows past M read as zero)
  {
    const float4* A4 = (const float4*)A;
    float4* As4 = (float4*)As;
    for (int i = tid; i < ROWS * KV; i += 256) {
      int r = i / KV;
      int c = i - r * KV;
      int gr = row0 + r;
      float4 v = {0.f, 0.f, 0.f, 0.f};
      if (gr < M) v = A4[(size_t)gr * KV + c];
      As4[i] = v;
    }
  }
  __syncthreads();

  const int mtile = wave / NT;
  const int ntile = wave - mtile * NT;
  const int lrow = (mtile << 4) + (lane & 15);   // A row inside LDS tile
  const int koff = (lane >> 4) << 1;             // 0 for lanes 0-15, 2 for 16-31
  const int ncol = (ntile << 4) + (lane & 15);   // B/C column
  const int bbase = ((lane >> 4) * NCOL + ncol); // float2 index into Bp

  v8f acc = {0.f, 0.f, 0.f, 0.f, 0.f, 0.f, 0.f, 0.f};
  for (int kk = 0; kk < KV; ++kk) {
    v2f a, b;
    a.x = As[lrow * K + kk * 4 + koff];
    a.y = As[lrow * K + kk * 4 + koff + 1];
    b = *(const v2f*)&Bp[(size_t)(bbase + kk * 2 * NCOL) * 2];
    acc = __builtin_amdgcn_wmma_f32_16x16x4_f32(
        /*neg_a=*/false, a, /*neg_b=*/false, b,
        /*c_mod=*/(short)0, acc, /*reuse_a=*/false, /*reuse_b=*/false);
  }

  const int rbase = row0 + (mtile << 4) + ((lane >> 4) << 3);
  if (row0 + ROWS <= M) {
    // full tile: unguarded stores (block-uniform branch, EXEC stays all-1s)
#pragma unroll
    for (int i = 0; i < 8; ++i) C[(size_t)(rbase + i) * NCOL + ncol] = acc[i];
  } else {
#pragma unroll
    for (int i = 0; i < 8; ++i) {
      int r = rbase + i;
      if (r < M) C[(size_t)r * NCOL + ncol] = acc[i];
    }
  }
}

// ---------------- per-node attention coefficients ---------------------------
// block = H*F threads; alpha_s[n,h] = sum_f h[n,h,f]*a_s[h,f] (same for dst).
template <int H, int F>
__global__ void alpha_kernel(const float* __restrict__ h,
                             const float* __restrict__ a_s,
                             const float* __restrict__ a_d,
                             float* __restrict__ out_s,
                             float* __restrict__ out_d) {
  constexpr int HF = H * F;
  __shared__ float ss[HF];
  __shared__ float sd[HF];
  const int n = blockIdx.x;
  const int t = threadIdx.x;
  float v = h[(size_t)n * HF + t];
  ss[t] = v * a_s[t];
  sd[t] = v * a_d[t];
  __syncthreads();
  for (int stride = F / 2; stride > 0; stride >>= 1) {
    if ((t & (F - 1)) < stride) {
      ss[t] += ss[t + stride];
      sd[t] += sd[t + stride];
    }
    __syncthreads();
  }
  if ((t & (F - 1)) == 0) {
    int hd = t / F;
    out_s[n * H + hd] = ss[t];
    out_d[n * H + hd] = sd[t];
  }
}

// ---------------- edge pass 1: leaky-relu logits + segment max --------------
template <int H>
__global__ void edge_logits_kernel(const float* __restrict__ as,
                                   const float* __restrict__ ad,
                                   const int* __restrict__ src,
                                   const int* __restrict__ dst, int nE, int nTot,
                                   float* __restrict__ logits,
                                   unsigned* __restrict__ mbuf) {
  int tid = blockIdx.x * blockDim.x + threadIdx.x;
  if (tid >= nTot * H) return;
  int e = tid / H, hd = tid - e * H;
  int s, d;
  if (e < nE) { s = src[e]; d = dst[e]; } else { s = d = e - nE; }
  float v = as[s * H + hd] + ad[d * H + hd];
  v = v > 0.f ? v : 0.2f * v;  // leaky_relu(0.2)
  logits[tid] = v;
  atomicMax(&mbuf[d * H + hd], f2ord(v));
}

// ---------------- edge pass 2: softmax denominator --------------------------
template <int H>
__global__ void edge_denom_kernel(const float* __restrict__ logits,
                                  const unsigned* __restrict__ mbuf,
                                  const int* __restrict__ dst, int nE, int nTot,
                                  float* __restrict__ denom) {
  int tid = blockIdx.x * blockDim.x + threadIdx.x;
  if (tid >= nTot * H) return;
  int e = tid / H, hd = tid - e * H;
  int d = (e < nE) ? dst[e] : (e - nE);
  float m = ord2f(mbuf[d * H + hd]);
  atomicAdd(&denom[d * H + hd], __expf(logits[tid] - m));
}

// ---------------- edge pass 3: weighted scatter of source features ----------
// One thread handles 4 features (float4 gather, 4 scalar float atomics).
template <int H, int F>
__global__ void edge_agg_kernel(const float* __restrict__ hsrc,
                                const float* __restrict__ logits,
                                const unsigned* __restrict__ mbuf,
                                const float* __restrict__ denom,
                                const int* __restrict__ src,
                                const int* __restrict__ dst, int nE, int nTot,
                                float* __restrict__ agg) {
  constexpr int HF = H * F;
  constexpr int CH = HF / 4;
  int tid = blockIdx.x * blockDim.x + threadIdx.x;
  if (tid >= nTot * CH) return;
  int e = tid / CH;
  int c = tid - e * CH;
  int fbase = c * 4;
  int hd = fbase / F;
  int s, d;
  if (e < nE) { s = src[e]; d = dst[e]; } else { s = d = e - nE; }
  float m = ord2f(mbuf[d * H + hd]);
  float alpha = __expf(logits[e * H + hd] - m) / (denom[d * H + hd] + 1e-16f);
  const float4 hv = *(const float4*)&hsrc[(size_t)s * HF + fbase];
  float* ap = &agg[(size_t)d * HF + fbase];
  atomicAdd(ap + 0, hv.x * alpha);
  atomicAdd(ap + 1, hv.y * alpha);
  atomicAdd(ap + 2, hv.z * alpha);
  atomicAdd(ap + 3, hv.w * alpha);
}

// ---------------- bias + ELU (in place) -------------------------------------
__global__ void elu_bias_kernel(float* __restrict__ buf,
                                const float* __restrict__ bias, int total,
                                int ncol) {
  int tid = blockIdx.x * blockDim.x + threadIdx.x;
  if (tid >= total) return;
  float v = buf[tid] + bias[tid % ncol];
  buf[tid] = v > 0.f ? v : (__expf(v) - 1.f);
}

// ---------------- bias + log_softmax over 32 classes (1 node per wave) ------
__global__ __launch_bounds__(256) void logsoftmax_kernel(
    const float* __restrict__ agg, const float* __restrict__ bias,
    float* __restrict__ out, int n) {
  int tid = blockIdx.x * blockDim.x + threadIdx.x;
  int node = tid >> 5;
  int lane = tid & 31;
  if (node >= n) return;  // wave-uniform exit (32 lanes == 1 node)
  float v = agg[(size_t)node * 32 + lane] + bias[lane];
  float m = v;
  for (int o = 16; o > 0; o >>= 1) m = fmaxf(m, __shfl_xor(m, o));
  float ex = __expf(v - m);
  float s = ex;
  for (int o = 16; o > 0; o >>= 1) s += __shfl_xor(s, o);
  out[(size_t)node * 32 + lane] = v - m - __logf(s);
}

// ---------------------------------------------------------------------------
extern "C" void kernel_launch(void* const* d_in, const int* in_sizes, int n_in,
                              void* d_out, int out_size, void* d_ws,
                              size_t ws_size, hipStream_t stream) {
  (void)n_in; (void)out_size; (void)ws_size;
  const float* x      = (const float*)d_in[0];
  const int*   ei     = (const int*)d_in[1];
  const float* W1     = (const float*)d_in[2];
  const float* a_src1 = (const float*)d_in[3];
  const float* a_dst1 = (const float*)d_in[4];
  const float* b1     = (const float*)d_in[5];
  const float* W2     = (const float*)d_in[6];
  const float* a_src2 = (const float*)d_in[7];
  const float* a_dst2 = (const float*)d_in[8];
  const float* b2     = (const float*)d_in[9];

  const int n    = in_sizes[0] / 500;  // 50000 nodes
  const int nE   = in_sizes[1] / 2;    // 800000 edges
  const int nTot = nE + n;             // + self loops
  const int* src = ei;
  const int* dst = ei + nE;

  // ---- workspace layout (all 4-byte elems; ~78 MB total) ----
  float* ws = (float*)d_ws;
  size_t o = 0;
  float* h1      = ws + o; o += (size_t)n * 128;
  float* logits1 = ws + o; o += (size_t)nTot * 2;
  float* as1     = ws + o; o += (size_t)n * 2;
  float* ad1     = ws + o; o += (size_t)n * 2;
  float* h2      = ws + o; o += (size_t)n * 32;
  float* logits2 = ws + o; o += (size_t)nTot;
  float* as2     = ws + o; o += (size_t)n;
  float* ad2     = ws + o; o += (size_t)n;
  float* Bp1     = ws + o; o += (size_t)500 * 128;  // packed W1
  float* Bp2     = ws + o; o += (size_t)128 * 32;   // packed W2
  size_t z0 = o;  // start of zero-initialized region
  unsigned* m1   = (unsigned*)(ws + o); o += (size_t)n * 2;
  float* denom1  = ws + o; o += (size_t)n * 2;
  float* agg1    = ws + o; o += (size_t)n * 128;  // becomes h1' after ELU
  unsigned* m2   = (unsigned*)(ws + o); o += (size_t)n;
  float* denom2  = ws + o; o += (size_t)n;
  float* agg2    = ws + o; o += (size_t)n * 32;
  size_t zcount = o - z0;

  // 0) zero max/denom/agg buffers (m sentinel 0 == ordered minimum)
  zero_kernel<<<2048, 256, 0, stream>>>((unsigned*)(ws + z0), (long long)zcount);
  // 0b) pack both weight matrices into WMMA fragment order
  pack_b_kernel<500, 128><<<(500 * 128 + 255) / 256, 256, 0, stream>>>(W1, Bp1);
  pack_b_kernel<128, 32><<<(128 * 32 + 255) / 256, 256, 0, stream>>>(W2, Bp2);

  // ---- layer 1 ----
  gemm_wmma_kernel<500, 128><<<(n + 15) / 16, 256, 0, stream>>>(x, Bp1, h1, n);
  alpha_kernel<2, 64><<<n, 128, 0, stream>>>(h1, a_src1, a_dst1, as1, ad1);
  {
    int t = nTot * 2;
    edge_logits_kernel<2><<<(t + 255) / 256, 256, 0, stream>>>(
        as1, ad1, src, dst, nE, nTot, logits1, m1);
    edge_denom_kernel<2><<<(t + 255) / 256, 256, 0, stream>>>(
        logits1, m1, dst, nE, nTot, denom1);
    int ta = nTot * 32;  // 4 features per thread, 128 feats/edge
    edge_agg_kernel<2, 64><<<(ta + 255) / 256, 256, 0, stream>>>(
        h1, logits1, m1, denom1, src, dst, nE, nTot, agg1);
  }
  {
    int t = n * 128;
    elu_bias_kernel<<<(t + 255) / 256, 256, 0, stream>>>(agg1, b1, t, 128);
  }

  // ---- layer 2 ----
  gemm_wmma_kernel<128, 32><<<(n + 63) / 64, 256, 0, stream>>>(agg1, Bp2, h2, n);
  alpha_kernel<1, 32><<<n, 32, 0, stream>>>(h2, a_src2, a_dst2, as2, ad2);
  {
    int t = nTot;
    edge_logits_kernel<1><<<(t + 255) / 256, 256, 0, stream>>>(
        as2, ad2, src, dst, nE, nTot, logits2, m2);
    edge_denom_kernel<1><<<(t + 255) / 256, 256, 0, stream>>>(
        logits2, m2, dst, nE, nTot, denom2);
    int ta = nTot * 8;  // 4 features per thread, 32 feats/edge
    edge_agg_kernel<1, 32><<<(ta + 255) / 256, 256, 0, stream>>>(
        h2, logits2, m2, denom2, src, dst, nE, nTot, agg2);
  }

  // ---- output ----
  logsoftmax_kernel<<<((size_t)n * 32 + 255) / 256, 256, 0, stream>>>(
      agg2, b2, (float*)d_out, n);
}